// Conv2D_25726854103363
// MI455X (gfx1250) — compile-verified
//
#include <hip/hip_runtime.h>

typedef __attribute__((ext_vector_type(2))) float v2f;
typedef __attribute__((ext_vector_type(8))) float v8f;

#define H 4096
#define W 4096
#define KH 7
#define KW 7
#define OH (H - KH + 1)   /* 4090 */
#define OW (W - KW + 1)   /* 4090 */

#define TILE 16
#define ROWTILES 2                     /* vertical 16-row tiles per wave        */
#define BLK_ROWS (TILE * ROWTILES)     /* 32 output rows per block              */
#define BLK_COLS 128                   /* 8 waves x 16 output cols              */
#define LDS_ROWS (BLK_ROWS + KH - 1)   /* 38                                    */
#define LDS_COLS 136                   /* 128 + taps + K-pad (34 float4s)       */
#define LDS_COLS4 (LDS_COLS / 4)       /* 34                                    */
#define LDS_STRIDE 140                 /* 16B-aligned rows, bank-conflict-free  */
#define NCHUNK (6 * KH)                /* 42 K-chunks (t,i)                     */

__global__ __launch_bounds__(256)
void conv7x7_wmma_f32(const float* __restrict__ x, const float* __restrict__ w,
                      const float* __restrict__ bias, float* __restrict__ out)
{
    __shared__ __align__(16) float sX[LDS_ROWS * LDS_STRIDE]; /* input patch   */
    __shared__ float sB[NCHUNK * 32 * 2];        /* banded weight fragments     */

    const int tid  = threadIdx.x;
    const int r0   = blockIdx.y * BLK_ROWS;
    const int cblk = blockIdx.x * BLK_COLS;

    /* ---- one-time: build banded B fragments (lane-layout of V_WMMA B operand) ---- */
    for (int idx = tid; idx < NCHUNK * 32; idx += 256) {
        const int chunk = idx >> 5;            /* 0..41  */
        const int L     = idx & 31;            /* lane   */
        const int t     = chunk / KH;
        const int i     = chunk - t * KH;
        const int n     = L & 15;              /* B column (N index)            */
        const int kh    = (L >> 4) << 1;       /* lanes>=16 hold K+2/K+3        */
        const int k0    = 4 * t + kh;
        const int j0    = k0 - n;              /* filter tap for element 0      */
        const int j1    = j0 + 1;
        const float m0  = ((unsigned)j0 <= 6u) ? 1.0f : 0.0f;
        const float m1  = ((unsigned)j1 <= 6u) ? 1.0f : 0.0f;
        const int ja    = ((unsigned)j0 <= 6u) ? j0 : 0;   /* clamped, always valid */
        const int jb    = ((unsigned)j1 <= 6u) ? j1 : 0;
        v2f b;
        b.x = w[i * KW + ja] * m0;
        b.y = w[i * KW + jb] * m1;
        *(v2f*)&sB[(chunk * 32 + L) * 2] = b;
    }

    /* ---- fill the shared input patch ---- */
    const bool interior = (r0 + LDS_ROWS <= H) && (cblk + LDS_COLS <= W);
    if (interior) {
#if defined(__HIP_DEVICE_COMPILE__)
        /* CDNA5 async DMA: global -> LDS, 16B per lane, no VGPR staging. */
        const unsigned ldsBase = (unsigned)(size_t)(void*)sX;
        for (int idx = tid; idx < LDS_ROWS * LDS_COLS4; idx += 256) {
            const int row = idx / LDS_COLS4;
            const int c4  = idx - row * LDS_COLS4;
            const unsigned long long gaddr =
                (unsigned long long)(size_t)(x + (size_t)(r0 + row) * W + cblk + c4 * 4);
            const unsigned laddr = ldsBase + (unsigned)((row * LDS_STRIDE + c4 * 4) * 4);
            asm volatile("global_load_async_to_lds_b128 %0, %1, off"
                         :: "v"(laddr), "v"(gaddr) : "memory");
        }
        asm volatile("s_wait_asynccnt 0" ::: "memory");
#endif
    } else {
        /* edge blocks: scalar, index-clamped fallback */
        for (int idx = tid; idx < LDS_ROWS * LDS_COLS; idx += 256) {
            int row = idx / LDS_COLS;
            int col = idx - row * LDS_COLS;
            int gr = r0 + row;   if (gr > H - 1) gr = H - 1;
            int gc = cblk + col; if (gc > W - 1) gc = W - 1;
            sX[row * LDS_STRIDE + col] = x[gr * W + gc];
        }
    }
    __syncthreads();

    const int wave = tid >> 5;
    const int lane = tid & 31;
    const int m    = lane & 15;                /* A row within tile             */
    const int kh   = (lane >> 4) << 1;
    const int c0l  = wave * TILE;

    v8f acc0 = (v8f){0.f, 0.f, 0.f, 0.f, 0.f, 0.f, 0.f, 0.f};
    v8f acc1 = (v8f){0.f, 0.f, 0.f, 0.f, 0.f, 0.f, 0.f, 0.f};

    /* Out(32x16) = sum_i  X_i(32x24) * Band_i(24x16), K in 6 chunks of 4 */
    #pragma unroll
    for (int t = 0; t < 6; ++t) {
        const int abase = c0l + 4 * t + kh;
        #pragma unroll
        for (int i = 0; i < KH; ++i) {
            v2f b  = *(const v2f*)&sB[((t * KH + i) * 32 + lane) * 2];
            v2f a0 = *(const v2f*)&sX[(i + m)        * LDS_STRIDE + abase];
            v2f a1 = *(const v2f*)&sX[(TILE + i + m) * LDS_STRIDE + abase];
            acc0 = __builtin_amdgcn_wmma_f32_16x16x4_f32(
                       false, a0, false, b, (short)0, acc0, false, false);
            acc1 = __builtin_amdgcn_wmma_f32_16x16x4_f32(
                       false, a1, false, b, (short)0, acc1, false, false);
        }
    }

    /* ---- store: C layout = VGPR v holds M = v + (lane<16 ? 0 : 8), N = lane%16 ---- */
    const float bv  = bias[0];
    const int mbase = (lane >> 4) * 8;
    const int ocol  = cblk + c0l + m;
    if (ocol < OW) {
        #pragma unroll
        for (int v = 0; v < 8; ++v) {
            int orow0 = r0 + mbase + v;
            int orow1 = orow0 + TILE;
            if (orow0 < OH) out[orow0 * OW + ocol] = acc0[v] + bv;
            if (orow1 < OH) out[orow1 * OW + ocol] = acc1[v] + bv;
        }
    }
}

extern "C" void kernel_launch(void* const* d_in, const int* in_sizes, int n_in,
                              void* d_out, int out_size, void* d_ws, size_t ws_size,
                              hipStream_t stream)
{
    const float* x  = (const float*)d_in[0];
    const float* w  = (const float*)d_in[1];
    const float* b  = (const float*)d_in[2];
    float* out      = (float*)d_out;

    dim3 grid((OW + BLK_COLS - 1) / BLK_COLS, (OH + BLK_ROWS - 1) / BLK_ROWS);
    conv7x7_wmma_f32<<<grid, dim3(256), 0, stream>>>(x, w, b, out);
}